// clique_69329362092378
// MI455X (gfx1250) — compile-verified
//
#include <hip/hip_runtime.h>
#include <hip/hip_bf16.h>

// ---------------------------------------------------------------------------
// GNN forward for MI455X (gfx1250, wave32).
//
// Pipeline (all fp32, WMMA f32 16x16x4 for the two real GEMMs):
//   1. y  = x @ W0 + b0                       [N,128]   (WMMA)
//   2. column sum/sumsq of y -> GraphNorm affine (a,c) per feature
//   3. fold (a,c) into W0_2 -> x1 = relu(y @ W' + b')   [N,10]
//   4. ye = edge_attr @ W0e + b0e             [E,32]    (WMMA)
//   5. column stats of ye -> edge affine (aE,cE)
//   6. per layer l: fold (aE,cE,b_l) into Wq [32,10], bq [10]
//        p   = x @ W_l[0:128] + x1 @ W_l[128:138]        [N,10]
//        agg[dst] += p[src] + (ye @ Wq + bq)   (atomics)
//        x1  = relu(agg)
//   7. logits = x1 @ W5 + b5 ; softmax over pickable rows.
//
// GraphNorm fold: out = (y - ms*m)*rsqrt(var+eps)*w + b = y*a + c, with
//   m = sum/n, var = sumsq/n - m^2*ms*(2-ms), a = inv*w, c = b - ms*m*inv*w.
// ---------------------------------------------------------------------------

#define Nn   50000
#define Ee   800000
#define XF   128
#define EF   32
#define OUTC 2
#define NPK  20000
#define HID  10

typedef __attribute__((ext_vector_type(2))) float v2f;
typedef __attribute__((ext_vector_type(8))) float v8f;

// ---------------- WMMA GEMM: Y = A[nrows,K] @ W[K,ncols] + bias ------------
// 256 threads = 8 waves. Each wave owns one 16x16 output tile.
// ncols=128 -> 8 col tiles, 1 row tile/block. ncols=32 -> 2 col tiles, 4 row
// tiles/block. Grids chosen so every wave has a full tile (EXEC all ones).
__global__ void gemm_wmma(const float* __restrict__ A, const float* __restrict__ W,
                          const float* __restrict__ bias, float* __restrict__ Y,
                          int nrows, int K, int ncols) {
    int wave = threadIdx.x >> 5;
    int lane = threadIdx.x & 31;
    int colTiles = ncols >> 4;
    int rowTilesPerBlock = 8 / colTiles;
    int row0 = (blockIdx.x * rowTilesPerBlock + (wave / colTiles)) * 16;
    int col0 = (wave % colTiles) * 16;
    if (row0 >= nrows) return;           // wave-uniform (grids are exact anyway)

    int lr = lane & 15;                  // tile row (A/D) or tile col (B/D)
    int hi = lane >> 4;                  // half-wave select

    v8f acc = {};
    for (int k = 0; k < K; k += 4) {
        int ka = k + hi * 2;
        // A 16x4 f32 frag: lanes 0-15 -> (M=lr, K=k,k+1); lanes 16-31 -> K=k+2,k+3
        const float* ar = A + (size_t)(row0 + lr) * K + ka;
        v2f af; af.x = ar[0]; af.y = ar[1];
        // B 4x16 f32 frag: lanes 0-15 -> rows k,k+1; lanes 16-31 -> rows k+2,k+3
        v2f bf; bf.x = W[(size_t)ka * ncols + col0 + lr];
                bf.y = W[(size_t)(ka + 1) * ncols + col0 + lr];
        acc = __builtin_amdgcn_wmma_f32_16x16x4_f32(false, af, false, bf,
                                                    (short)0, acc, false, false);
    }
    // D layout: VGPR v: lanes 0-15 -> M=v, lanes 16-31 -> M=v+8; N=lr
    float bv = bias[col0 + lr];
    int rbase = row0 + hi * 8;
#pragma unroll
    for (int v = 0; v < 8; v++)
        Y[(size_t)(rbase + v) * ncols + col0 + lr] = acc[v] + bv;
}

// ---------------- column sum / sumsq (for GraphNorm) -----------------------
// blockDim = (C, 256/C); each block covers rowsPerBlock rows.
__global__ void colstats(const float* __restrict__ M, int nrows, int ncols,
                         int rowsPerBlock, float* __restrict__ sum,
                         float* __restrict__ sumsq) {
    int c = threadIdx.x, ty = threadIdx.y, R = blockDim.y;
    int r0 = blockIdx.x * rowsPerBlock;
    int r1 = r0 + rowsPerBlock; if (r1 > nrows) r1 = nrows;
    float s = 0.f, ss = 0.f;
    for (int r = r0 + ty; r < r1; r += R) {
        float v = M[(size_t)r * ncols + c];
        s += v; ss += v * v;
    }
    __shared__ float sm[512];
    int idx = ty * ncols + c;            // ncols*R == 256 always
    sm[idx] = s; sm[256 + idx] = ss;
    __syncthreads();
    if (ty == 0) {
        for (int t = 1; t < R; t++) { s += sm[t * ncols + c]; ss += sm[256 + t * ncols + c]; }
        atomicAdd(&sum[c], s);
        atomicAdd(&sumsq[c], ss);
    }
}

// ---------------- stats -> per-feature affine (a, c) -----------------------
__global__ void coeffs(const float* __restrict__ sum, const float* __restrict__ sumsq,
                       const float* __restrict__ w, const float* __restrict__ b,
                       const float* __restrict__ ms, float invn,
                       float* __restrict__ a, float* __restrict__ c) {
    int f = threadIdx.x;
    float m   = sum[f] * invn;
    float msf = ms[f];
    float var = sumsq[f] * invn - m * m * msf * (2.0f - msf);
    float inv = rsqrtf(var + 1e-5f);
    a[f] = inv * w[f];
    c[f] = b[f] - msf * m * inv * w[f];
}

// ---------------- fold affine into a [K,J] weight --------------------------
// Wout[k][j] = a[k]*W[k][j];  bout[j] = b[j] + sum_k c[k]*W[k][j]
__global__ void fold_weights(const float* __restrict__ a, const float* __restrict__ c,
                             const float* __restrict__ W, const float* __restrict__ b,
                             int K, int J, float* __restrict__ Wout, float* __restrict__ bout) {
    for (int i = threadIdx.x; i < K * J; i += blockDim.x)
        Wout[i] = a[i / J] * W[i];
    if (threadIdx.x < J) {
        float s = b[threadIdx.x];
        for (int k = 0; k < K; k++) s += c[k] * W[k * J + threadIdx.x];
        bout[threadIdx.x] = s;
    }
}

// ---------------- x1 = relu(Y @ Wf + bf), K=128, J=10 ----------------------
__global__ void map_rows_relu(const float* __restrict__ Y, const float* __restrict__ Wf,
                              const float* __restrict__ bf, float* __restrict__ Xout,
                              int nrows) {
    __shared__ float sw[XF * HID];
    __shared__ float sb[HID];
    for (int i = threadIdx.x; i < XF * HID; i += blockDim.x) sw[i] = Wf[i];
    if (threadIdx.x < HID) sb[threadIdx.x] = bf[threadIdx.x];
    __syncthreads();
    int r = blockIdx.x * blockDim.x + threadIdx.x;
    if (r >= nrows) return;
    float acc[HID];
#pragma unroll
    for (int j = 0; j < HID; j++) acc[j] = sb[j];
    const float* yr = Y + (size_t)r * XF;
    for (int k = 0; k < XF; k++) {
        float v = yr[k];
#pragma unroll
        for (int j = 0; j < HID; j++) acc[j] += v * sw[k * HID + j];
    }
    float* xo = Xout + (size_t)r * HID;
#pragma unroll
    for (int j = 0; j < HID; j++) xo[j] = fmaxf(acc[j], 0.0f);
}

// ---------------- p = x @ W_l[0:128] + x1 @ W_l[128:138] -------------------
__global__ void node_p(const float* __restrict__ X, const float* __restrict__ X1,
                       const float* __restrict__ Wl, float* __restrict__ P, int nrows) {
    __shared__ float sw[(XF + HID) * HID];   // 1380 floats
    for (int i = threadIdx.x; i < (XF + HID) * HID; i += blockDim.x) sw[i] = Wl[i];
    __syncthreads();
    int r = blockIdx.x * blockDim.x + threadIdx.x;
    if (r >= nrows) return;
    float acc[HID] = {};
    const float* xr = X + (size_t)r * XF;
    for (int k = 0; k < XF; k++) {
        float v = xr[k];
#pragma unroll
        for (int j = 0; j < HID; j++) acc[j] += v * sw[k * HID + j];
    }
    const float* x1r = X1 + (size_t)r * HID;
#pragma unroll
    for (int k = 0; k < HID; k++) {
        float v = x1r[k];
#pragma unroll
        for (int j = 0; j < HID; j++) acc[j] += v * sw[(XF + k) * HID + j];
    }
    float* pr = P + (size_t)r * HID;
#pragma unroll
    for (int j = 0; j < HID; j++) pr[j] = acc[j];
}

// ---------------- edge scatter: agg[dst] += p[src] + ye@Wq + bq ------------
__global__ void scatter_edges(const int* __restrict__ src, const int* __restrict__ dst,
                              const float* __restrict__ YE, const float* __restrict__ Wq,
                              const float* __restrict__ bq, const float* __restrict__ P,
                              float* __restrict__ Agg, int nedges) {
    __shared__ float sw[EF * HID];
    __shared__ float sb[HID];
    for (int i = threadIdx.x; i < EF * HID; i += blockDim.x) sw[i] = Wq[i];
    if (threadIdx.x < HID) sb[threadIdx.x] = bq[threadIdx.x];
    __syncthreads();
    int e = blockIdx.x * blockDim.x + threadIdx.x;
    if (e >= nedges) return;
    int s = src[e], d = dst[e];
    float acc[HID];
#pragma unroll
    for (int j = 0; j < HID; j++) acc[j] = sb[j];
    const float* yr = YE + (size_t)e * EF;
#pragma unroll
    for (int k = 0; k < EF; k++) {
        float v = yr[k];
#pragma unroll
        for (int j = 0; j < HID; j++) acc[j] += v * sw[k * HID + j];
    }
    const float* ps = P + (size_t)s * HID;
    float* ad = Agg + (size_t)d * HID;
#pragma unroll
    for (int j = 0; j < HID; j++) atomicAdd(&ad[j], acc[j] + ps[j]);
}

__global__ void relu_inplace(float* __restrict__ a, int n) {
    int i = blockIdx.x * blockDim.x + threadIdx.x;
    if (i < n) a[i] = fmaxf(a[i], 0.0f);
}

// ---------------- logits + softmax over pickable ---------------------------
__global__ void final_softmax(const float* __restrict__ X1, const float* __restrict__ W5,
                              const float* __restrict__ b5, const int* __restrict__ pick,
                              float* __restrict__ out, int npick) {
    int i = blockIdx.x * blockDim.x + threadIdx.x;
    if (i >= npick) return;
    int n = pick[i];
    const float* xr = X1 + (size_t)n * HID;
    float l0 = b5[0], l1 = b5[1];
#pragma unroll
    for (int j = 0; j < HID; j++) {
        float v = xr[j];
        l0 += v * W5[j * 2 + 0];
        l1 += v * W5[j * 2 + 1];
    }
    float m = fmaxf(l0, l1);
    float e0 = expf(l0 - m), e1 = expf(l1 - m);
    float inv = 1.0f / (e0 + e1);
    out[i * 2 + 0] = e0 * inv;
    out[i * 2 + 1] = e1 * inv;
}

// ---------------------------------------------------------------------------
extern "C" void kernel_launch(void* const* d_in, const int* in_sizes, int n_in,
                              void* d_out, int out_size, void* d_ws, size_t ws_size,
                              hipStream_t stream) {
    const float* x         = (const float*)d_in[0];
    const int*   edge_idx  = (const int*)d_in[2];
    const int*   src       = edge_idx;
    const int*   dst       = edge_idx + Ee;
    const float* edge_attr = (const float*)d_in[6];
    const int*   pickable  = (const int*)d_in[7];
    const float* W0   = (const float*)d_in[8];
    const float* b0   = (const float*)d_in[9];
    const float* W0_2 = (const float*)d_in[10];
    const float* b0_2 = (const float*)d_in[11];
    const float* W0e  = (const float*)d_in[12];
    const float* b0e  = (const float*)d_in[13];
    const float* n_w  = (const float*)d_in[14];
    const float* n_b  = (const float*)d_in[15];
    const float* n_ms = (const float*)d_in[16];
    const float* n2_w  = (const float*)d_in[17];
    const float* n2_b  = (const float*)d_in[18];
    const float* n2_ms = (const float*)d_in[19];
    const float* Wl[4] = { (const float*)d_in[20], (const float*)d_in[22],
                           (const float*)d_in[24], (const float*)d_in[26] };
    const float* bl[4] = { (const float*)d_in[21], (const float*)d_in[23],
                           (const float*)d_in[25], (const float*)d_in[27] };
    const float* W5 = (const float*)d_in[28];
    const float* b5 = (const float*)d_in[29];
    float* out = (float*)d_out;

    // ---- workspace carve (~134 MB) ----
    float* ws = (float*)d_ws;
    size_t off = 0;
    float* y    = ws + off; off += (size_t)Nn * XF;     // 6.4 M
    float* ye   = ws + off; off += (size_t)Ee * EF;     // 25.6 M
    float* x1a  = ws + off; off += (size_t)Nn * HID;
    float* x1b  = ws + off; off += (size_t)Nn * HID;
    float* pbuf = ws + off; off += (size_t)Nn * HID;
    float* stats = ws + off; off += 320;                // sumN(128) ssqN(128) sumE(32) ssqE(32)
    float* sumN = stats, *ssqN = stats + 128, *sumE = stats + 256, *ssqE = stats + 288;
    float* aN = ws + off; off += 128;
    float* cN = ws + off; off += 128;
    float* aE = ws + off; off += 32;
    float* cE = ws + off; off += 32;
    float* wfold = ws + off; off += XF * HID;
    float* bfold = ws + off; off += 16;
    float* wq = ws + off; off += EF * HID;
    float* bq = ws + off; off += 16;

    hipMemsetAsync(stats, 0, 320 * sizeof(float), stream);

    // 1. y = x@W0 + b0  (WMMA f32): 3125 blocks * (1 row tile x 8 col tiles)
    gemm_wmma<<<Nn / 16, 256, 0, stream>>>(x, W0, b0, y, Nn, XF, XF);

    // 2. node GraphNorm stats -> affine
    colstats<<<(Nn + 511) / 512, dim3(128, 2), 0, stream>>>(y, Nn, 128, 512, sumN, ssqN);
    coeffs<<<1, 128, 0, stream>>>(sumN, ssqN, n_w, n_b, n_ms, 1.0f / Nn, aN, cN);

    // 3. fold into W0_2; x1 = relu(y @ W' + b')
    fold_weights<<<1, 256, 0, stream>>>(aN, cN, W0_2, b0_2, XF, HID, wfold, bfold);
    map_rows_relu<<<(Nn + 255) / 256, 256, 0, stream>>>(y, wfold, bfold, x1a, Nn);

    // 4. ye = edge_attr@W0e + b0e (WMMA): 12500 blocks * (4 row x 2 col tiles)
    gemm_wmma<<<Ee / 64, 256, 0, stream>>>(edge_attr, W0e, b0e, ye, Ee, EF, EF);

    // 5. edge GraphNorm stats -> affine
    colstats<<<(Ee + 511) / 512, dim3(32, 8), 0, stream>>>(ye, Ee, 32, 512, sumE, ssqE);
    coeffs<<<1, 32, 0, stream>>>(sumE, ssqE, n2_w, n2_b, n2_ms, 1.0f / Ee, aE, cE);

    // 6. four message-passing layers (ping-pong x1a <-> x1b)
    float* cur = x1a, *nxt = x1b;
    for (int l = 0; l < 4; l++) {
        // fold edge affine + bias into edge half of W_l (rows 138..169)
        fold_weights<<<1, 256, 0, stream>>>(aE, cE, Wl[l] + (XF + HID) * HID, bl[l],
                                            EF, HID, wq, bq);
        node_p<<<(Nn + 255) / 256, 256, 0, stream>>>(x, cur, Wl[l], pbuf, Nn);
        hipMemsetAsync(nxt, 0, (size_t)Nn * HID * sizeof(float), stream);
        scatter_edges<<<(Ee + 255) / 256, 256, 0, stream>>>(src, dst, ye, wq, bq,
                                                            pbuf, nxt, Ee);
        relu_inplace<<<(Nn * HID + 255) / 256, 256, 0, stream>>>(nxt, Nn * HID);
        float* t = cur; cur = nxt; nxt = t;
    }

    // 7. logits + softmax over pickable
    final_softmax<<<(NPK + 255) / 256, 256, 0, stream>>>(cur, W5, b5, pickable, out, NPK);
}